// AttentionLayer_85959475462289
// MI455X (gfx1250) — compile-verified
//
#include <hip/hip_runtime.h>
#include <cmath>

#define HID  768
#define GENE 512

typedef float v2f __attribute__((ext_vector_type(2)));
typedef float v8f __attribute__((ext_vector_type(8)));

// ---------------------------------------------------------------------------
// C[M,N] (+)= act(A[M,K] @ op(W) + bias), fp32 via V_WMMA_F32_16X16X4_F32.
// WT=1: W stored [N,K] row-major, use W^T (PyTorch-style `@ w.T`)
// WT=0: W stored [K,N] row-major, use W
// One wave computes one 16x16 tile; M,N multiples of 16, K multiple of 4.
// Fragment layouts per CDNA5 ISA 7.12.2 (16x4 A / 4x16 B / 16x16 C f32).
// ---------------------------------------------------------------------------
template <int WT, int RELU, int ACCUM>
__global__ void wmma_gemm_f32(const float* __restrict__ A,
                              const float* __restrict__ W,
                              const float* __restrict__ bias,
                              float* __restrict__ C,
                              int M, int N, int K) {
  const int lane = threadIdx.x & 31;
  const int wave = threadIdx.x >> 5;
  const int tilesM = M >> 4;
  const int tile = blockIdx.x * (blockDim.x >> 5) + wave;
  if (tile >= tilesM * (N >> 4)) return;          // whole-wave uniform
  const int tm = (tile % tilesM) << 4;
  const int tn = (tile / tilesM) << 4;

  const int r  = lane & 15;                       // row (A/C) or col (B/C)
  const int hi = lane >> 4;

  v8f acc = {};
  const float* arow = A + (size_t)(tm + r) * K + 2 * hi;   // A[m][k0+2hi]
  for (int k0 = 0; k0 < K; k0 += 4) {
    v2f a, b;
    a.x = arow[k0];
    a.y = arow[k0 + 1];
    if (WT) {                                     // B[k][n] = W[n][k]
      const float* wr = W + (size_t)(tn + r) * K + k0 + 2 * hi;
      b.x = wr[0];
      b.y = wr[1];
    } else {                                      // B[k][n] = W[k][n]
      const float* wc = W + (size_t)(k0 + 2 * hi) * N + tn + r;
      b.x = wc[0];
      b.y = wc[N];
    }
    acc = __builtin_amdgcn_wmma_f32_16x16x4_f32(false, a, false, b,
                                                (short)0, acc, false, false);
  }

  const float bv = bias ? bias[tn + r] : 0.0f;
#pragma unroll
  for (int v = 0; v < 8; ++v) {                   // C: vgpr v -> row v + 8*hi
    const int m = tm + v + 8 * hi;
    float out = acc[v] + bv;
    if (RELU) out = fmaxf(out, 0.0f);
    const size_t idx = (size_t)m * N + tn + r;
    if (ACCUM) C[idx] += out; else C[idx] = out;
  }
}

// ---------------------------------------------------------------------------
// c[b] = Q[b,:]·kb  for both layers. One wave per (graph, layer).
// ---------------------------------------------------------------------------
__global__ void cdot_kernel(const float* __restrict__ Q0,
                            const float* __restrict__ Q1,
                            const float* __restrict__ kb0,
                            const float* __restrict__ kb1,
                            float* __restrict__ c0, float* __restrict__ c1,
                            int B) {
  const int lane = threadIdx.x & 31;
  const int wave = threadIdx.x >> 5;
  const int idx = blockIdx.x * (blockDim.x >> 5) + wave;
  if (idx >= 2 * B) return;
  const int b = idx >> 1, layer = idx & 1;
  const float* q  = (layer ? Q1 : Q0) + (size_t)b * HID;
  const float* kb = layer ? kb1 : kb0;
  float d = 0.0f;
  for (int j = lane; j < HID; j += 32) d += q[j] * kb[j];
#pragma unroll
  for (int off = 16; off; off >>= 1) d += __shfl_xor(d, off, 32);
  if (lane == 0) (layer ? c1 : c0)[b] = d;
}

// ---------------------------------------------------------------------------
// Streaming pass 1: scores for both layers. One wave per node.
// s[i] = (q~[g]·x[i] + c[g]) / sqrt(HID),  g = batch_ids[i].
// ---------------------------------------------------------------------------
__global__ void scores_kernel(const float* __restrict__ x,
                              const int* __restrict__ bids,
                              const float* __restrict__ qt0,
                              const float* __restrict__ qt1,
                              const float* __restrict__ c0,
                              const float* __restrict__ c1,
                              float* __restrict__ s0,
                              float* __restrict__ s1,
                              int N, float rscale) {
  const int lane = threadIdx.x & 31;
  const int wave = threadIdx.x >> 5;
  const int node = blockIdx.x * (blockDim.x >> 5) + wave;
  if (node >= N) return;
  const int g = bids[node];
  const float4* xv = (const float4*)(x + (size_t)node * HID);
  const float4* q0 = (const float4*)(qt0 + (size_t)g * HID);
  const float4* q1 = (const float4*)(qt1 + (size_t)g * HID);
  float d0 = 0.0f, d1 = 0.0f;
#pragma unroll
  for (int j = 0; j < HID / 128; ++j) {           // 6 x float4 per lane
    const int idx = j * 32 + lane;
    const float4 xx = xv[idx];
    const float4 a0 = q0[idx];
    const float4 a1 = q1[idx];
    d0 += xx.x * a0.x + xx.y * a0.y + xx.z * a0.z + xx.w * a0.w;
    d1 += xx.x * a1.x + xx.y * a1.y + xx.z * a1.z + xx.w * a1.w;
  }
#pragma unroll
  for (int off = 16; off; off >>= 1) {
    d0 += __shfl_xor(d0, off, 32);
    d1 += __shfl_xor(d1, off, 32);
  }
  if (lane == 0) {
    s0[node] = (d0 + c0[g]) * rscale;
    s1[node] = (d1 + c1[g]) * rscale;
  }
}

// ---------------------------------------------------------------------------
// Streaming pass 2: per-graph softmax + weighted sum S[b] = Σ attn_j x_j.
// One 256-thread block per graph; thread t owns columns {t, t+256, t+512}.
// batch_ids is sorted -> graph node range found by binary search.
// ---------------------------------------------------------------------------
__global__ void graph_pool_kernel(const float* __restrict__ x,
                                  const int* __restrict__ bids,
                                  const float* __restrict__ s0,
                                  const float* __restrict__ s1,
                                  float* __restrict__ S0,
                                  float* __restrict__ S1,
                                  int N) {
  const int b = blockIdx.x;
  const int t = threadIdx.x;
  __shared__ float r0[256];
  __shared__ float r1[256];

  int lo = 0, hi = N;                             // first idx with bids >= b
  while (lo < hi) { int m = (lo + hi) >> 1; if (bids[m] < b) lo = m + 1; else hi = m; }
  const int start = lo;
  hi = N;                                         // first idx with bids >= b+1
  while (lo < hi) { int m = (lo + hi) >> 1; if (bids[m] < b + 1) lo = m + 1; else hi = m; }
  const int end = lo;

  float acc0[3] = {0.f, 0.f, 0.f}, acc1[3] = {0.f, 0.f, 0.f};
  if (start < end) {
    float m0 = -1e30f, m1 = -1e30f;
    for (int j = start + t; j < end; j += 256) {
      m0 = fmaxf(m0, s0[j]);
      m1 = fmaxf(m1, s1[j]);
    }
    r0[t] = m0; r1[t] = m1;
    __syncthreads();
    for (int off = 128; off; off >>= 1) {
      if (t < off) { r0[t] = fmaxf(r0[t], r0[t + off]); r1[t] = fmaxf(r1[t], r1[t + off]); }
      __syncthreads();
    }
    m0 = r0[0]; m1 = r1[0];
    __syncthreads();

    float l0 = 0.f, l1 = 0.f;
    for (int j = start + t; j < end; j += 256) {
      l0 += __expf(s0[j] - m0);
      l1 += __expf(s1[j] - m1);
    }
    r0[t] = l0; r1[t] = l1;
    __syncthreads();
    for (int off = 128; off; off >>= 1) {
      if (t < off) { r0[t] += r0[t + off]; r1[t] += r1[t + off]; }
      __syncthreads();
    }
    const float inv0 = 1.0f / r0[0], inv1 = 1.0f / r1[0];

    for (int j = start; j < end; ++j) {           // x row read is coalesced
      const float w0 = __expf(s0[j] - m0) * inv0; // scalar: cache broadcast
      const float w1 = __expf(s1[j] - m1) * inv1;
      const float* xr = x + (size_t)j * HID;
#pragma unroll
      for (int k = 0; k < 3; ++k) {
        const float xv = xr[t + k * 256];
        acc0[k] += w0 * xv;
        acc1[k] += w1 * xv;
      }
    }
  }
#pragma unroll
  for (int k = 0; k < 3; ++k) {
    S0[(size_t)b * HID + t + k * 256] = acc0[k];
    S1[(size_t)b * HID + t + k * 256] = acc1[k];
  }
}

// ---------------------------------------------------------------------------
extern "C" void kernel_launch(void* const* d_in, const int* in_sizes, int n_in,
                              void* d_out, int out_size, void* d_ws, size_t ws_size,
                              hipStream_t stream) {
  const float* x      = (const float*)d_in[0];
  const int*   bids   = (const int*)  d_in[1];
  const float* gene   = (const float*)d_in[2];
  const float* bionic = (const float*)d_in[3];
  const float* fc0_w  = (const float*)d_in[6];
  const float* fc0_b  = (const float*)d_in[7];
  const float* fc1_w  = (const float*)d_in[8];
  const float* fc1_b  = (const float*)d_in[9];
  const float* a0_qw = (const float*)d_in[10], *a0_qb = (const float*)d_in[11];
  const float* a0_kw = (const float*)d_in[12], *a0_kb = (const float*)d_in[13];
  const float* a0_vw = (const float*)d_in[14], *a0_vb = (const float*)d_in[15];
  const float* a0_ow = (const float*)d_in[16], *a0_ob = (const float*)d_in[17];
  const float* a1_qw = (const float*)d_in[18], *a1_qb = (const float*)d_in[19];
  const float* a1_kw = (const float*)d_in[20], *a1_kb = (const float*)d_in[21];
  const float* a1_vw = (const float*)d_in[22], *a1_vb = (const float*)d_in[23];
  const float* a1_ow = (const float*)d_in[24], *a1_ob = (const float*)d_in[25];

  const int N = in_sizes[1];                // 49152 nodes
  const int B = in_sizes[2] / GENE;         // 128 graphs

  // workspace layout (floats)
  float* ws  = (float*)d_ws;
  const size_t BH = (size_t)B * HID;
  float* G   = ws + 0 * BH;
  float* Bio = ws + 1 * BH;
  float* Q0  = ws + 2 * BH;
  float* Q1  = ws + 3 * BH;
  float* Qt0 = ws + 4 * BH;
  float* Qt1 = ws + 5 * BH;
  float* S0  = ws + 6 * BH;
  float* S1  = ws + 7 * BH;
  float* T0  = ws + 8 * BH;
  float* T1  = ws + 9 * BH;
  float* c0  = ws + 10 * BH;
  float* c1  = c0 + B;
  float* sc0 = c1 + B;
  float* sc1 = sc0 + N;

  const int gemm_tiles  = (B / 16) * (HID / 16);        // 384
  const int gemm_blocks = (gemm_tiles + 7) / 8;         // 8 waves/block
  const dim3 blk(256);

  // 1) g = relu(gene @ fc0^T + b), bio = relu(bionic @ fc1^T + b)
  wmma_gemm_f32<1,1,0><<<gemm_blocks, blk, 0, stream>>>(gene,   fc0_w, fc0_b, G,   B, HID, GENE);
  wmma_gemm_f32<1,1,0><<<gemm_blocks, blk, 0, stream>>>(bionic, fc1_w, fc1_b, Bio, B, HID, GENE);

  // 2) Q = g @ qw^T + qb
  wmma_gemm_f32<1,0,0><<<gemm_blocks, blk, 0, stream>>>(G,   a0_qw, a0_qb, Q0, B, HID, HID);
  wmma_gemm_f32<1,0,0><<<gemm_blocks, blk, 0, stream>>>(Bio, a1_qw, a1_qb, Q1, B, HID, HID);

  // 3) q~ = Q @ kw   (K-projection folded into the query)
  wmma_gemm_f32<0,0,0><<<gemm_blocks, blk, 0, stream>>>(Q0, a0_kw, nullptr, Qt0, B, HID, HID);
  wmma_gemm_f32<0,0,0><<<gemm_blocks, blk, 0, stream>>>(Q1, a1_kw, nullptr, Qt1, B, HID, HID);

  // 4) c = Q·kb
  cdot_kernel<<<(2 * B + 7) / 8, blk, 0, stream>>>(Q0, Q1, a0_kb, a1_kb, c0, c1, B);

  // 5) scores: one streaming pass over x (151 MB)
  const float rscale = 1.0f / sqrtf((float)HID);
  scores_kernel<<<(N + 7) / 8, blk, 0, stream>>>(x, bids, Qt0, Qt1, c0, c1, sc0, sc1, N, rscale);

  // 6) softmax + weighted node sum (second x pass, mostly L2-resident)
  graph_pool_kernel<<<B, blk, 0, stream>>>(x, bids, sc0, sc1, S0, S1, N);

  // 7) out = ow @ (vw @ S + vb) + ob, summed over both layers
  wmma_gemm_f32<1,0,0><<<gemm_blocks, blk, 0, stream>>>(S0, a0_vw, a0_vb, T0, B, HID, HID);
  wmma_gemm_f32<1,0,0><<<gemm_blocks, blk, 0, stream>>>(S1, a1_vw, a1_vb, T1, B, HID, HID);
  wmma_gemm_f32<1,0,0><<<gemm_blocks, blk, 0, stream>>>(T0, a0_ow, a0_ob, (float*)d_out, B, HID, HID);
  wmma_gemm_f32<1,0,1><<<gemm_blocks, blk, 0, stream>>>(T1, a1_ow, a1_ob, (float*)d_out, B, HID, HID);
}